// KarplusSynth_61959198212121
// MI455X (gfx1250) — compile-verified
//
#include <hip/hip_runtime.h>
#include <math.h>

// ---------------- problem constants ----------------
#define SRF      44100.0f
#define NSAMP    8388608
#define W        882          // wavetable size
#define NB       9510         // full KS blocks  (9510*882 = 8387820)
#define REM      788          // remainder samples
#define L        256          // band-reject block length
#define P        32768        // number of blocks (P*L == NSAMP)
#define CH       128          // scan chunk size
#define NCH      256          // P/CH

// ---------------- workspace layout (float offsets) ----------------
#define WT_OFF   0            // 882   filtered wavetable
#define U_OFF    1024         // L     homogeneous response u[n]
#define V_OFF    1280         // L     homogeneous response v[n]
#define PW_OFF   1536         // (CH+1)*4  powers of T (row-major 2x2)
#define BX_OFF   2560         // 2*P   boundary x pairs
#define FS_OFF   68096        // 2*P   block final particular states
#define IS_OFF   133632      // 2*P   block initial states (p,q)
#define CS_OFF   199168       // 2*NCH chunk offset sums
#define CSS_OFF  199680       // 2*NCH chunk start states

typedef float v2f __attribute__((ext_vector_type(2)));
typedef float v8f __attribute__((ext_vector_type(8)));

__device__ __forceinline__ float clipf(float x, float lo, float hi) {
    return fminf(fmaxf(x, lo), hi);
}

// normalized band-reject coefficients: c[0..4] = b0n,b1n,b2n,a1n,a2n
__device__ __forceinline__ void br_coeffs(const float* h, float* c) {
    float fr = clipf(h[5] * SRF / 4.0f, 100.0f, SRF / 2.0f - 1.0f);
    float q  = clipf(h[6], 0.1f, 0.999f);
    float w0 = 2.0f * 3.14159265358979f * fr / SRF;
    float alpha = sinf(w0) / (2.0f * q);
    float cw = cosf(w0);
    float inv_a0 = 1.0f / (1.0f + alpha);
    c[0] = inv_a0;                    // b0/a0
    c[1] = -2.0f * cw * inv_a0;       // b1/a0
    c[2] = inv_a0;                    // b2/a0
    c[3] = -2.0f * cw * inv_a0;       // a1/a0
    c[4] = (1.0f - alpha) * inv_a0;   // a2/a0
}

__device__ __forceinline__ void run_biquad_inplace(const float* x, float* y, int n,
                                                   float b0, float b1, float b2,
                                                   float a0, float a1, float a2) {
    float ia = 1.0f / a0;
    float b0n = b0 * ia, b1n = b1 * ia, b2n = b2 * ia;
    float a1n = a1 * ia, a2n = a2 * ia;
    float x1 = 0.f, x2 = 0.f, y1 = 0.f, y2 = 0.f;
    for (int i = 0; i < n; ++i) {
        float xn = x[i];
        float yn = b0n * xn + b1n * x1 + b2n * x2 - a1n * y1 - a2n * y2;
        y[i] = yn;
        x2 = x1; x1 = xn; y2 = y1; y1 = yn;
    }
}

// ---------------- kernel A: tiny serial prep ----------------
__global__ void prep_kernel(const float* __restrict__ wavetable,
                            const float* __restrict__ h,
                            const float* __restrict__ lp2,
                            float* __restrict__ ws) {
    int t = threadIdx.x;
    if (t == 0) {
        float lpf = clipf(h[1] * SRF / 4.0f, 100.0f, SRF / 2.0f - 1.0f);
        float lpq = clipf(h[2], 0.1f, 0.999f);
        float w0 = 2.0f * 3.14159265358979f * lpf / SRF;
        float alpha = sinf(w0) / (2.0f * lpq);
        float cw = cosf(w0);
        float b0 = (1.0f - cw) * 0.5f;
        run_biquad_inplace(wavetable, ws + WT_OFF, W,
                           b0, 1.0f - cw, b0, 1.0f + alpha, -2.0f * cw, 1.0f - alpha);
        float c2 = 100.0f + lp2[0] * 8000.0f;
        w0 = 2.0f * 3.14159265358979f * c2 / SRF;
        alpha = sinf(w0) / (2.0f * 0.707f);
        cw = cosf(w0);
        b0 = (1.0f - cw) * 0.5f;
        run_biquad_inplace(ws + WT_OFF, ws + WT_OFF, W,
                           b0, 1.0f - cw, b0, 1.0f + alpha, -2.0f * cw, 1.0f - alpha);
    } else if (t == 1) {
        float c[5]; br_coeffs(h, c);
        float u1 = 1.f, u2 = 0.f, v1 = 0.f, v2 = 1.f; // u[-1]=1,u[-2]=0 ; v[-1]=0,v[-2]=1
        for (int i = 0; i < L; ++i) {
            float u = -c[3] * u1 - c[4] * u2;
            float v = -c[3] * v1 - c[4] * v2;
            ws[U_OFF + i] = u; ws[V_OFF + i] = v;
            u2 = u1; u1 = u; v2 = v1; v1 = v;
        }
        // T = [[u[L-1], v[L-1]], [u[L-2], v[L-2]]]; store T^0..T^CH
        float t00 = u1, t01 = v1, t10 = u2, t11 = v2;
        float m00 = 1.f, m01 = 0.f, m10 = 0.f, m11 = 1.f;
        for (int k = 0; k <= CH; ++k) {
            ws[PW_OFF + 4 * k + 0] = m00; ws[PW_OFF + 4 * k + 1] = m01;
            ws[PW_OFF + 4 * k + 2] = m10; ws[PW_OFF + 4 * k + 3] = m11;
            float n00 = t00 * m00 + t01 * m10;
            float n01 = t00 * m01 + t01 * m11;
            float n10 = t10 * m00 + t11 * m10;
            float n11 = t10 * m01 + t11 * m11;
            m00 = n00; m01 = n01; m10 = n10; m11 = n11;
        }
    }
}

// ---------------- kernel B: Karplus-Strong persistent workgroup ----------------
// 896 threads (28 wave32s). One barrier per step; LDS double-buffered for the
// roll. Feedback load is software-pipelined one step ahead (register) and the
// L2 is warmed 8 steps ahead via global_prefetch_b8, so the serial chain is
// just tanh -> LDS write -> barrier -> LDS read -> fma.
__global__ __launch_bounds__(896) void ks_kernel(const float* __restrict__ fb,
                                                 const float* __restrict__ h,
                                                 const float* __restrict__ ws,
                                                 float* __restrict__ out) {
    __shared__ float buf[2][W];
    int t = threadIdx.x;
    float decay = clipf(h[0] * 0.1f + 0.9f, 0.9f, 0.999f);
    float amt = h[3];
    float cw = (t < W) ? ws[WT_OFF + t] : 0.0f;
    float fcur = (t < W) ? fb[t] : 0.0f;
    for (int k = 0; k < NB; ++k) {
        float cwp = cw;
        float fnext = 0.0f;
        if (t < W) {
            if (k + 1 < NB) fnext = fb[(k + 1) * W + t];            // pipelined load
            if (k + 8 < NB) __builtin_prefetch(&fb[(k + 8) * W + t], 0, 1);
            cwp = cw + tanhf(amt * fcur);
            buf[k & 1][t] = cwp;
        }
        __syncthreads();
        if (t < W) {
            float left = buf[k & 1][(t + W - 1) % W];
            cw = decay * 0.5f * (cwp + left);
            out[k * W + t] = cw;
        }
        fcur = fnext;
        // no second barrier needed: next step writes the other buffer
    }
    // remainder block: smoothing only, then fade on the final 256 samples
    {
        float cwp = cw;
        if (t < W) buf[NB & 1][t] = cwp;
        __syncthreads();
        if (t < W && t < REM) {
            float left = buf[NB & 1][(t + W - 1) % W];
            float val = decay * 0.5f * (cwp + left);
            int tail = t - (REM - 256);
            if (tail >= 0) val *= (1.0f - (float)tail * (1.0f / 255.0f));
            out[NB * W + t] = val;
        }
    }
}

// ---------------- kernel C0: stage boundary x pairs (avoid in-place race) ----
__global__ void bx_kernel(const float* __restrict__ out, float* __restrict__ ws) {
    int j = blockIdx.x * blockDim.x + threadIdx.x;
    if (j >= P) return;
    ws[BX_OFF + 2 * j]     = (j > 0) ? out[j * L - 1] : 0.0f;
    ws[BX_OFF + 2 * j + 1] = (j > 0) ? out[j * L - 2] : 0.0f;
}

// ---------------- kernel C1: per-block particular solution (in place) -------
__global__ void br_part_kernel(const float* __restrict__ h,
                               float* __restrict__ out,
                               float* __restrict__ ws) {
    int j = blockIdx.x * blockDim.x + threadIdx.x;
    if (j >= P) return;
    float c[5]; br_coeffs(h, c);
    float x1 = ws[BX_OFF + 2 * j], x2 = ws[BX_OFF + 2 * j + 1];
    float y1 = 0.f, y2 = 0.f;
    int base = j * L;
    for (int i = 0; i < L; ++i) {
        float xn = out[base + i];
        float yn = c[0] * xn + c[1] * x1 + c[2] * x2 - c[3] * y1 - c[4] * y2;
        out[base + i] = yn;
        x2 = x1; x1 = xn; y2 = y1; y1 = yn;
    }
    ws[FS_OFF + 2 * j] = y1;       // y[L-1]
    ws[FS_OFF + 2 * j + 1] = y2;   // y[L-2]
}

// ---------------- kernel C2a: local affine scan inside each chunk -----------
__global__ void scan_local_kernel(float* __restrict__ ws) {
    int c = threadIdx.x;
    if (c >= NCH) return;
    float t00 = ws[U_OFF + L - 1], t01 = ws[V_OFF + L - 1];
    float t10 = ws[U_OFF + L - 2], t11 = ws[V_OFF + L - 2];
    float p = 0.f, q = 0.f;
    int base = c * CH;
    for (int i = 0; i < CH; ++i) {
        ws[IS_OFF + 2 * (base + i)] = p;          // local offset o_i
        ws[IS_OFF + 2 * (base + i) + 1] = q;
        float f1 = ws[FS_OFF + 2 * (base + i)];
        float f2 = ws[FS_OFF + 2 * (base + i) + 1];
        float np = t00 * p + t01 * q + f1;
        float nq = t10 * p + t11 * q + f2;
        p = np; q = nq;
    }
    ws[CS_OFF + 2 * c] = p; ws[CS_OFF + 2 * c + 1] = q;
}

// ---------------- kernel C2b: 256-element serial scan over chunks -----------
__global__ void scan_chunks_kernel(float* __restrict__ ws) {
    if (threadIdx.x != 0 || blockIdx.x != 0) return;
    float m00 = ws[PW_OFF + 4 * CH + 0], m01 = ws[PW_OFF + 4 * CH + 1];
    float m10 = ws[PW_OFF + 4 * CH + 2], m11 = ws[PW_OFF + 4 * CH + 3];
    float p = 0.f, q = 0.f;
    for (int c = 0; c < NCH; ++c) {
        ws[CSS_OFF + 2 * c] = p; ws[CSS_OFF + 2 * c + 1] = q;
        float np = m00 * p + m01 * q + ws[CS_OFF + 2 * c];
        float nq = m10 * p + m11 * q + ws[CS_OFF + 2 * c + 1];
        p = np; q = nq;
    }
}

// ---------------- kernel C2c: fixup per-block init states -------------------
__global__ void scan_fix_kernel(float* __restrict__ ws) {
    int j = blockIdx.x * blockDim.x + threadIdx.x;
    if (j >= P) return;
    int c = j / CH, i = j % CH;
    float m00 = ws[PW_OFF + 4 * i + 0], m01 = ws[PW_OFF + 4 * i + 1];
    float m10 = ws[PW_OFF + 4 * i + 2], m11 = ws[PW_OFF + 4 * i + 3];
    float sp = ws[CSS_OFF + 2 * c], sq = ws[CSS_OFF + 2 * c + 1];
    float op = ws[IS_OFF + 2 * j], oq = ws[IS_OFF + 2 * j + 1];
    ws[IS_OFF + 2 * j]     = m00 * sp + m01 * sq + op;
    ws[IS_OFF + 2 * j + 1] = m10 * sp + m11 * sq + oq;
}

// ---------------- kernel C3: WMMA rank-2 correction + envelope --------------
// Each wave owns one 16(pos) x 16(block) tile:
//   D = A(16x4) * B(4x16) + C ;  A=[u,v,0,0] rows, B=[p;q;0;0] cols, C=y_part
// f32 16x16x4 VGPR layout per ISA: A in 2 VGPRs (lanes0-15: K0/K1; lanes16-31:
// K2/K3 = zeros); B/C/D row-striped (VGPR r: lanes0-15 row r, lanes16-31 row r+8).
// C/D tile moved as 2x b128 per lane (base is 32B-aligned). Envelope uses
// hoisted reciprocals to avoid per-element IEEE divide chains.
__global__ __launch_bounds__(256) void correct_env_kernel(const float* __restrict__ h,
                                                          const float* __restrict__ envp,
                                                          const float* __restrict__ ws,
                                                          float* __restrict__ out) {
    int lane = threadIdx.x & 31;
    int wave = threadIdx.x >> 5;
    int tile = blockIdx.x * 8 + wave;          // 4096 blocks * 8 waves = 32768 tiles
    int posTile = tile & 15;                   // L/16 = 16 position tiles
    int blkTile = tile >> 4;                   // P/16 = 2048 block tiles
    int posBase = posTile * 16;
    int blkBase = blkTile * 16;

    float ax = 0.f, ay = 0.f, bx = 0.f, by = 0.f;
    if (lane < 16) {
        ax = ws[U_OFF + posBase + lane];       // A[M=lane][K=0] = u[n]
        ay = ws[V_OFF + posBase + lane];       // A[M=lane][K=1] = v[n]
        int j = blkBase + lane;
        bx = ws[IS_OFF + 2 * j];               // B[K=0][N=lane] = p_j
        by = ws[IS_OFF + 2 * j + 1];           // B[K=1][N=lane] = q_j
    }
    v2f A = {ax, ay};
    v2f B = {bx, by};

    int col  = blkBase + (lane & 15);
    int mOff = (lane >= 16) ? 8 : 0;
    int base = col * L + posBase + mOff;       // global sample index of D[r=0]

    const float4* cp = (const float4*)(out + base);
    float4 ca = cp[0];
    float4 cb = cp[1];
    v8f C = {ca.x, ca.y, ca.z, ca.w, cb.x, cb.y, cb.z, cb.w};

    v8f D = __builtin_amdgcn_wmma_f32_16x16x4_f32(
        false, A, false, B, (short)0, C, false, false);

    float n = (float)NSAMP;
    float inv_attack  = 1.0f / (1.0f + envp[0] * 0.1f * n);
    float inv_release = 1.0f / (1.0f + envp[2] * 0.1f * n);
    float sa = clipf(envp[1], 0.f, 1.f) * h[4];   // sustain * amplitude

    float e[8];
#pragma unroll
    for (int r = 0; r < 8; ++r) {
        float g = (float)(base + r);
        e[r] = clipf(g * inv_attack, 0.f, 1.f) *
               clipf((n - 1.0f - g) * inv_release, 0.f, 1.f) * sa;
    }
    float4 oa = make_float4(D[0] * e[0], D[1] * e[1], D[2] * e[2], D[3] * e[3]);
    float4 ob = make_float4(D[4] * e[4], D[5] * e[5], D[6] * e[6], D[7] * e[7]);
    float4* op = (float4*)(out + base);
    op[0] = oa;
    op[1] = ob;
}

// ---------------- launcher ----------------
extern "C" void kernel_launch(void* const* d_in, const int* in_sizes, int n_in,
                              void* d_out, int out_size, void* d_ws, size_t ws_size,
                              hipStream_t stream) {
    (void)in_sizes; (void)n_in; (void)out_size; (void)ws_size;
    const float* fb   = (const float*)d_in[0];
    const float* wt   = (const float*)d_in[1];
    const float* h    = (const float*)d_in[2];
    const float* envp = (const float*)d_in[3];
    const float* lp2  = (const float*)d_in[4];
    float* out = (float*)d_out;
    float* ws  = (float*)d_ws;

    prep_kernel       <<<1, 64, 0, stream>>>(wt, h, lp2, ws);
    ks_kernel         <<<1, 896, 0, stream>>>(fb, h, ws, out);
    bx_kernel         <<<P / 256, 256, 0, stream>>>(out, ws);
    br_part_kernel    <<<P / 256, 256, 0, stream>>>(h, out, ws);
    scan_local_kernel <<<1, 256, 0, stream>>>(ws);
    scan_chunks_kernel<<<1, 32, 0, stream>>>(ws);
    scan_fix_kernel   <<<P / 256, 256, 0, stream>>>(ws);
    correct_env_kernel<<<(16 * (P / 16)) / 8, 256, 0, stream>>>(h, envp, ws, out);
}